// HashEmbedder_33560874451696
// MI455X (gfx1250) — compile-verified
//
#include <hip/hip_runtime.h>

// HashEmbedder (Instant-NGP hash grid) forward for MI455X / gfx1250.
// Memory-bound gather kernel: 64M random 8B gathers into 64MB of tables
// (L2-resident on MI455X's 192MB L2), NT stores for the 64MB write-once
// output, async global->LDS staging of the point tile (gfx1250 ASYNCcnt
// path), 8 batched independent b64 gathers per level for latency hiding.

typedef float vf2 __attribute__((ext_vector_type(2)));

#define NLVL  16
#define TSZ   (1u << 19)
#define TMASK (TSZ - 1u)
#define P0    73856093u
#define P1    19349663u
#define P2    83492791u
#define BLK   256

__global__ __launch_bounds__(BLK) void hashgrid_fwd(
    const float* __restrict__ pts,   // [npts,3]
    const float* __restrict__ tab,   // [16, 2^19, 2]
    float* __restrict__ out,         // [npts, 32]
    int npts)
{
    __shared__ float spts[BLK * 3];   // 3 KB: this block's point tile
    const int t    = threadIdx.x;
    const int base = blockIdx.x * BLK;

    if (base + BLK <= npts) {
        // Stage 768 dwords (256 points x 3 floats) into LDS with the CDNA5
        // async copy engine. INST_OFFSET is added to both LDS and global
        // addresses, so one (lds,global) pair + 3 offsets covers the tile.
        unsigned     ldsa = (unsigned)(unsigned long long)&spts[t];
        const float* g    = pts + (size_t)base * 3 + t;
        asm volatile(
            "global_load_async_to_lds_b32 %0, %1, off\n\t"
            "global_load_async_to_lds_b32 %0, %1, off offset:1024\n\t"
            "global_load_async_to_lds_b32 %0, %1, off offset:2048\n\t"
            "s_wait_asynccnt 0x0"
            :: "v"(ldsa), "v"(g) : "memory");
    } else {
        // Partial tail block (not hit for B=524288): plain guarded loads.
        for (int i = t; i < BLK * 3; i += BLK) {
            int p = base * 3 + i;
            spts[i] = (p < npts * 3) ? pts[p] : 0.0f;
        }
    }
    __syncthreads();

    const int b = base + t;
    if (b >= npts) return;

    const float px = spts[3 * t + 0];
    const float py = spts[3 * t + 1];
    const float pz = spts[3 * t + 2];

    // grid = (BOX_MAX-BOX_MIN)/res with res = floor(16 * 2^(l/3)),
    // i.e. {16,20,25,32,40,50,64,80,101,128,161,203,256,322,406,512}.
    constexpr float GRIDS[NLVL] = {
        2.0f / 16.0f,  2.0f / 20.0f,  2.0f / 25.0f,  2.0f / 32.0f,
        2.0f / 40.0f,  2.0f / 50.0f,  2.0f / 64.0f,  2.0f / 80.0f,
        2.0f / 101.0f, 2.0f / 128.0f, 2.0f / 161.0f, 2.0f / 203.0f,
        2.0f / 256.0f, 2.0f / 322.0f, 2.0f / 406.0f, 2.0f / 512.0f };

    float* outp = out + (size_t)b * (2 * NLVL);

    #pragma unroll
    for (int l = 0; l < NLVL; ++l) {
        const float grid = GRIDS[l];
        const float sx = (px + 1.0f) / grid;
        const float sy = (py + 1.0f) / grid;
        const float sz = (pz + 1.0f) / grid;
        const float fx = floorf(sx), fy = floorf(sy), fz = floorf(sz);
        const float wx = sx - fx,   wy = sy - fy,   wz = sz - fz;

        // (x+1)*P == x*P + P (mod 2^32): 3 muls + 3 adds -> all 8 hashes.
        const unsigned x = (unsigned)fx, y = (unsigned)fy, z = (unsigned)fz;
        const unsigned hx0 = x * P0, hx1 = hx0 + P0;
        const unsigned hy0 = y * P1, hy1 = hy0 + P1;
        const unsigned hz0 = z * P2, hz1 = hz0 + P2;

        // Issue all 8 independent gathers (global_load_b64) up front.
        const vf2* __restrict__ tl = (const vf2*)tab + (size_t)l * TSZ;
        const vf2 e000 = tl[(hx0 ^ hy0 ^ hz0) & TMASK];
        const vf2 e001 = tl[(hx0 ^ hy0 ^ hz1) & TMASK];
        const vf2 e010 = tl[(hx0 ^ hy1 ^ hz0) & TMASK];
        const vf2 e011 = tl[(hx0 ^ hy1 ^ hz1) & TMASK];
        const vf2 e100 = tl[(hx1 ^ hy0 ^ hz0) & TMASK];
        const vf2 e101 = tl[(hx1 ^ hy0 ^ hz1) & TMASK];
        const vf2 e110 = tl[(hx1 ^ hy1 ^ hz0) & TMASK];
        const vf2 e111 = tl[(hx1 ^ hy1 ^ hz1) & TMASK];

        // Trilinear weights, corner order (i outer, j, k inner).
        const float ux = 1.0f - wx, uy = 1.0f - wy, uz = 1.0f - wz;
        const float c00 = ux * uy, c01 = ux * wy;
        const float c10 = wx * uy, c11 = wx * wy;
        const float w000 = c00 * uz, w001 = c00 * wz;
        const float w010 = c01 * uz, w011 = c01 * wz;
        const float w100 = c10 * uz, w101 = c10 * wz;
        const float w110 = c11 * uz, w111 = c11 * wz;

        vf2 acc = w000 * e000 + w001 * e001 + w010 * e010 + w011 * e011
                + w100 * e100 + w101 * e101 + w110 * e110 + w111 * e111;

        // Output is written once and never re-read: NT store keeps the
        // 64MB result from evicting the L2-resident tables.
        __builtin_nontemporal_store(acc, (vf2*)outp + l);
    }
}

extern "C" void kernel_launch(void* const* d_in, const int* in_sizes, int n_in,
                              void* d_out, int out_size, void* d_ws, size_t ws_size,
                              hipStream_t stream)
{
    const float* pts = (const float*)d_in[0];   // [B,3] f32
    const float* tab = (const float*)d_in[1];   // [16, 2^19, 2] f32
    float*       out = (float*)d_out;           // [B, 32] f32

    const int npts   = in_sizes[0] / 3;
    const int blocks = (npts + BLK - 1) / BLK;

    hipLaunchKernelGGL(hashgrid_fwd, dim3(blocks), dim3(BLK), 0, stream,
                       pts, tab, out, npts);
}